// TCR_78331613545253
// MI455X (gfx1250) — compile-verified
//
#include <hip/hip_runtime.h>
#include <cstdint>

// ---------------------------------------------------------------------------
// Affine warp (rotation/shear/translation, bilinear, zero padding) for
// img [B=32, C=3, 512, 512] f32.  Memory-bound gather: stage the per-tile
// source bounding box into LDS with CDNA5 async global->LDS copies (b128
// fast path on 16B-aligned interior groups), then bilinear-sample from LDS.
// Zero padding is materialized in the LDS tile.
// ---------------------------------------------------------------------------

#define TILE     32
#define BB       80     // max bbox rows/cols (worst ~75 + 3 alignment)
#define LSTRIDE  84     // multiple of 4 (16B-aligned rows); 84 % 64 = 20 banks
#define ANG_F    0.34906585039886590f   // 20 deg in rad
#define MAXT_F   6.0f

// --- CDNA5 async global->LDS copies, GV addressing mode --------------------
__device__ __forceinline__ void async_g2l_b32(const float* gp, uint32_t lds_byte_off) {
  asm volatile("global_load_async_to_lds_b32 %0, %1, off"
               :: "v"(lds_byte_off), "v"((uint64_t)(uintptr_t)gp)
               : "memory");
}
__device__ __forceinline__ void async_g2l_b128(const float* gp, uint32_t lds_byte_off) {
  asm volatile("global_load_async_to_lds_b128 %0, %1, off"
               :: "v"(lds_byte_off), "v"((uint64_t)(uintptr_t)gp)
               : "memory");
}
__device__ __forceinline__ void wait_async0() {
#if __has_builtin(__builtin_amdgcn_s_wait_asynccnt)
  __builtin_amdgcn_s_wait_asynccnt(0);
#else
  asm volatile("s_wait_asynccnt 0x0" ::: "memory");
#endif
}

// --- per-sample inverse affine params (mirrors the reference exactly) ------
__global__ __launch_bounds__(64) void TCR_affine_params(
    const float* __restrict__ rnd, float* __restrict__ P,
    int Wp, int Hp, int B) {
  int b = blockIdx.x * blockDim.x + threadIdx.x;
  if (b >= B) return;
  float u  = rnd[b];
  float tx = 2.f * MAXT_F * u - MAXT_F;
  float ty = 2.f * MAXT_F * u - MAXT_F;
  float r  = 2.f * ANG_F  * u - ANG_F;
  float hx = 2.f * ANG_F  * u - ANG_F;
  float hy = 2.f * ANG_F  * u - ANG_F;
  float a  = hx - r;
  float bb = hy + r;
  float ca = cosf(a),  sa = sinf(a);
  float cb = cosf(bb), sb = sinf(bb);
  float chx = cosf(hx), chy = cosf(hy);
  float W = (float)Wp, H = (float)Hp;                 // W = Hin, H = Win (ref naming)
  float T11 = ca / chx;
  float T12 = sa / chx;
  float T13 = (W*chx - W*ca + 2.f*tx*ca - H*sa + 2.f*ty*sa) / (2.f*chx);
  float T21 = sb / chy;
  float T22 = cb / chy;
  float T23 = (H*chy - W*cb + 2.f*ty*cb - W*sb + 2.f*tx*sb) / (2.f*chy);
  float det = T11*T22 - T12*T21;
  float i00 =  T22/det, i01 = -T12/det;
  float i10 = -T21/det, i11 =  T11/det;
  float i02 = -(i00*T13 + i01*T23);
  float i12 = -(i10*T13 + i11*T23);
  float* p = P + b * 8;
  p[0]=i00; p[1]=i01; p[2]=i02; p[3]=i10; p[4]=i11; p[5]=i12; p[6]=0.f; p[7]=0.f;
}

// --- main warp kernel: one 32x32 output tile per 256-thread block ----------
__global__ __launch_bounds__(256) void TCR_warp_affine(
    const float* __restrict__ img, const float* __restrict__ P,
    float* __restrict__ out, int C, int Hin, int Win, int h_out, int w_out) {
  __shared__ __align__(16) float tile[BB * LSTRIDE];

  const int tilesX = w_out / TILE;
  const int tx0 = (blockIdx.x % tilesX) * TILE;
  const int ty0 = (blockIdx.x / tilesX) * TILE;
  const int plane = blockIdx.y;             // b*C + c
  const int b = plane / C;

  const float* Pb = P + b * 8;              // uniform -> scalar loads
  const float i00 = Pb[0], i01 = Pb[1], i02 = Pb[2];
  const float i10 = Pb[3], i11 = Pb[4], i12 = Pb[5];

  // Source bbox of this tile from the 4 corners (affine -> extrema at corners)
  const float X0 = (float)tx0, X1 = (float)(tx0 + TILE - 1);
  const float Y0 = (float)ty0, Y1 = (float)(ty0 + TILE - 1);
  float sxmin = i02 + fminf(i00*X0, i00*X1) + fminf(i01*Y0, i01*Y1);
  float sxmax = i02 + fmaxf(i00*X0, i00*X1) + fmaxf(i01*Y0, i01*Y1);
  float symin = i12 + fminf(i10*X0, i10*X1) + fminf(i11*Y0, i11*Y1);
  float symax = i12 + fmaxf(i10*X0, i10*X1) + fmaxf(i11*Y0, i11*Y1);

  // -1 margin guards fmaf-vs-separate rounding; align left edge to 4 elems
  // so interior groups are 16B-aligned in both global memory and LDS.
  const int fx0 = ((int)floorf(sxmin) - 1) & ~3;
  const int fy0 = (int)floorf(symin) - 1;
  int bw = (int)floorf(sxmax) + 3 - fx0;  if (bw > BB) bw = BB;
  int bh = (int)floorf(symax) + 3 - fy0;  if (bh > BB) bh = BB;

  const float* src = img + (size_t)plane * (size_t)Hin * (size_t)Win;

  const int t    = threadIdx.x;
  const int lane = t & 31;
  const int wv   = t >> 5;                  // 0..7 (8 wave32s)

  {
    int pr = fy0 < 0 ? 0 : fy0;
    __builtin_prefetch(src + (size_t)pr * Win, 0, 0);   // global_prefetch_b8
  }

  // Stage bbox into LDS: b128 async on fully-in-image aligned groups,
  // b32 async / zero-fill on edge groups.  ngx <= 20, so one wave pass/row.
  const int ngx = (bw + 3) >> 2;
  for (int j = wv; j < bh; j += 8) {
    const int gy = fy0 + j;
    const bool rowIn = (gy >= 0) && (gy < Hin);
    const uint32_t rowOff = (uint32_t)(uintptr_t)&tile[j * LSTRIDE];
    const float* grow = src + (size_t)gy * (size_t)Win;
    for (int g = lane; g < ngx; g += 32) {
      const int i4 = g << 2;
      const int gx = fx0 + i4;
      if (rowIn && gx >= 0 && gx + 3 < Win && i4 + 3 < bw) {
        async_g2l_b128(grow + gx, rowOff + 16u * (uint32_t)g);
      } else if (!rowIn || gx + 3 < 0 || gx >= Win) {
        // whole group outside the image (or dead row): vector zero-fill
        *(float4*)&tile[j * LSTRIDE + i4] = make_float4(0.f, 0.f, 0.f, 0.f);
      } else {
#pragma unroll
        for (int e = 0; e < 4; ++e) {
          const int i = i4 + e;
          if (i >= bw) break;
          const int gxx = gx + e;
          if (gxx >= 0 && gxx < Win) async_g2l_b32(grow + gxx, rowOff + 4u * (uint32_t)i);
          else                       tile[j * LSTRIDE + i] = 0.f;
        }
      }
    }
  }
  wait_async0();
  __syncthreads();

  // Bilinear sample from LDS; all taps fall inside the staged bbox, and
  // out-of-image bbox cells hold 0 -> zero padding handled for free.
  float* dst = out + (size_t)plane * (size_t)h_out * (size_t)w_out;
  const int x = tx0 + lane;
  const float fx = (float)x;
#pragma unroll
  for (int k = 0; k < 4; ++k) {
    const int y = ty0 + wv + k * 8;
    if (x >= w_out || y >= h_out) continue;
    const float fy = (float)y;
    const float sx = fmaf(i00, fx, fmaf(i01, fy, i02));
    const float sy = fmaf(i10, fx, fmaf(i11, fy, i12));
    const float x0 = floorf(sx), y0 = floorf(sy);
    const float wx1 = sx - x0, wy1 = sy - y0;
    const float wx0 = 1.f - wx1, wy0 = 1.f - wy1;
    const int xi = (int)x0 - fx0;
    const int yi = (int)y0 - fy0;
    const float* trow = &tile[yi * LSTRIDE + xi];
    const float v00 = trow[0];
    const float v01 = trow[1];
    const float v10 = trow[LSTRIDE];
    const float v11 = trow[LSTRIDE + 1];
    const float res = v00 * (wy0 * wx0) + v01 * (wy0 * wx1)
                    + v10 * (wy1 * wx0) + v11 * (wy1 * wx1);
    __builtin_nontemporal_store(res, &dst[(size_t)y * (size_t)w_out + x]);
  }
}

extern "C" void kernel_launch(void* const* d_in, const int* in_sizes, int n_in,
                              void* d_out, int out_size, void* d_ws, size_t ws_size,
                              hipStream_t stream) {
  const float* img = (const float*)d_in[0];
  const float* rnd = (const float*)d_in[1];
  float* out = (float*)d_out;
  float* P   = (float*)d_ws;          // 32 * 8 floats of scratch

  const int B   = in_sizes[1];        // 32
  const int Hin = 512, Win = 512;
  const int C   = in_sizes[0] / (B * Hin * Win);   // 3
  const int h_out = Hin, w_out = Win; // dsize=(W,H) with W=Hin, H=Win

  TCR_affine_params<<<1, 64, 0, stream>>>(rnd, P, /*W=*/Hin, /*H=*/Win, B);

  dim3 grid((w_out / TILE) * (h_out / TILE), B * C);
  TCR_warp_affine<<<grid, 256, 0, stream>>>(img, P, out, C, Hin, Win, h_out, w_out);
}